// DepthNet_weighted_5858335392005
// MI455X (gfx1250) — compile-verified
//
#include <hip/hip_runtime.h>
#include <hip/hip_bf16.h>
#include <math.h>

// ---------------------------------------------------------------------------
// MVS depth net forward for MI455X (gfx1250, wave32).
// Conv2d 3x3 layers run as implicit-im2col GEMM on V_WMMA_F32_16X16X4_F32
// (fp32 matrix pipe, keeps reference precision). A-fragments come from an
// LDS-staged input patch (bank-conflict-padded), B-fragments from a packed
// weight layout giving one coalesced b64 load per lane per WMMA. Fully
// unrolled K loop -> long runs of back-to-back WMMAs.
// ---------------------------------------------------------------------------

typedef float v2f __attribute__((ext_vector_type(2)));
typedef float v8f __attribute__((ext_vector_type(8)));

#define Hc   128
#define Wc   160
#define HWc  (Hc * Wc)          // 20480
#define Cc   32
#define Dc   64
#define Gc   8
#define Vc   4

// ---------------------------------------------------------------------------
// 1) Tiny projection kernel: new_proj for all views, inv(ref), rot/trans per
//    source view. One thread.
// ---------------------------------------------------------------------------
__global__ void proj_kernel(const float* __restrict__ pm /* (1,4,2,4,4) */,
                            float* __restrict__ rt /* 3 x 12 */) {
  if (threadIdx.x != 0 || blockIdx.x != 0) return;
  float np[Vc][4][4];
  for (int v = 0; v < Vc; ++v) {
    const float* E = pm + (v * 2 + 0) * 16;
    const float* K = pm + (v * 2 + 1) * 16;
    for (int i = 0; i < 4; ++i)
      for (int j = 0; j < 4; ++j) np[v][i][j] = E[i * 4 + j];
    for (int i = 0; i < 3; ++i)
      for (int j = 0; j < 4; ++j) {
        float s = 0.f;
        for (int k = 0; k < 3; ++k) s += K[i * 4 + k] * E[k * 4 + j];
        np[v][i][j] = s;
      }
  }
  // Gauss-Jordan inverse of np[0]
  float a[4][8];
  for (int i = 0; i < 4; ++i) {
    for (int j = 0; j < 4; ++j) { a[i][j] = np[0][i][j]; a[i][4 + j] = (i == j) ? 1.f : 0.f; }
  }
  for (int c = 0; c < 4; ++c) {
    int piv = c; float best = fabsf(a[c][c]);
    for (int r = c + 1; r < 4; ++r) { float m = fabsf(a[r][c]); if (m > best) { best = m; piv = r; } }
    if (piv != c) for (int j = 0; j < 8; ++j) { float t = a[c][j]; a[c][j] = a[piv][j]; a[piv][j] = t; }
    float inv = 1.f / a[c][c];
    for (int j = 0; j < 8; ++j) a[c][j] *= inv;
    for (int r = 0; r < 4; ++r) if (r != c) {
      float f = a[r][c];
      for (int j = 0; j < 8; ++j) a[r][j] -= f * a[c][j];
    }
  }
  for (int v = 1; v < Vc; ++v) {
    float M[3][4];
    for (int i = 0; i < 3; ++i)
      for (int j = 0; j < 4; ++j) {
        float s = 0.f;
        for (int k = 0; k < 4; ++k) s += np[v][i][k] * a[k][4 + j];
        M[i][j] = s;
      }
    float* o = rt + (v - 1) * 12;
    o[0] = M[0][0]; o[1] = M[0][1]; o[2] = M[0][2];
    o[3] = M[1][0]; o[4] = M[1][1]; o[5] = M[1][2];
    o[6] = M[2][0]; o[7] = M[2][1]; o[8] = M[2][2];
    o[9] = M[0][3]; o[10] = M[1][3]; o[11] = M[2][3];
  }
}

// ---------------------------------------------------------------------------
// 2) Init accumulators.
// ---------------------------------------------------------------------------
__global__ void init_kernel(float* __restrict__ vsum, float* __restrict__ wsum) {
  int t = blockIdx.x * blockDim.x + threadIdx.x;
  if (t < Gc * Dc * HWc) vsum[t] = 0.f;
  if (t < HWc) wsum[t] = 1e-5f;
}

// ---------------------------------------------------------------------------
// 3) Pack conv weights (OIHW) into WMMA-B-friendly layout:
//    k = (ky*3+kx)*CIN + ci ;  Bp[(k>>1)*128 + n*2 + (k&1)] = W[n][ci][ky][kx]
//    -> each lane's (K=k, k+1) B pair is one contiguous b64, rows of 128
//       floats are fully coalesced across the 16 lanes of a half-wave.
// ---------------------------------------------------------------------------
__global__ void pack_weights_kernel(const float* __restrict__ wgt, // 64 x CIN x 3 x 3
                                    float* __restrict__ bp, int cin) {
  int K = cin * 9;
  int t = blockIdx.x * blockDim.x + threadIdx.x;
  if (t >= 64 * K) return;
  int n = t / K;
  int k = t - n * K;
  int pos = k / cin;
  int ci = k - pos * cin;
  int ky = pos / 3, kx = pos - ky * 3;
  bp[(k >> 1) * 128 + n * 2 + (k & 1)] = wgt[((n * cin + ci) * 3 + ky) * 3 + kx];
}

// ---------------------------------------------------------------------------
// 4) Homography warp + bilinear sample + group correlation (per view).
// ---------------------------------------------------------------------------
__global__ void warp_gsim_kernel(const float* __restrict__ ref,   // 32 x HW
                                 const float* __restrict__ src,   // 32 x HW
                                 const float* __restrict__ dv,    // 64
                                 const float* __restrict__ rt,    // 12
                                 float* __restrict__ gsim,        // 8 x D x HW
                                 float* __restrict__ sim) {       // D x HW
  int t = blockIdx.x * blockDim.x + threadIdx.x;
  if (t >= Dc * HWc) return;
  int d = t / HWc;
  int p = t - d * HWc;
  int y = p / Wc, x = p - y * Wc;
  float dz = dv[d];
  float fxp = (float)x, fyp = (float)y;
  float rx = rt[0] * fxp + rt[1] * fyp + rt[2];
  float ry = rt[3] * fxp + rt[4] * fyp + rt[5];
  float rz = rt[6] * fxp + rt[7] * fyp + rt[8];
  float pz = rz * dz + rt[11];
  float px = (rx * dz + rt[9]) / pz;
  float py = (ry * dz + rt[10]) / pz;

  float x0f = floorf(px), y0f = floorf(py);
  float fx = px - x0f, fy = py - y0f;

  int   idx[4];
  float vld[4];
  float cx[2] = {x0f, x0f + 1.f};
  float cy[2] = {y0f, y0f + 1.f};
  #pragma unroll
  for (int cj = 0; cj < 2; ++cj) {
    #pragma unroll
    for (int ci = 0; ci < 2; ++ci) {
      float xi = cx[ci], yi = cy[cj];
      bool valid = (xi >= 0.f) && (xi <= (float)(Wc - 1)) &&
                   (yi >= 0.f) && (yi <= (float)(Hc - 1));
      float xc = fminf(fmaxf(xi, 0.f), (float)(Wc - 1));
      float yc = fminf(fmaxf(yi, 0.f), (float)(Hc - 1));
      idx[cj * 2 + ci] = (int)yc * Wc + (int)xc;
      vld[cj * 2 + ci] = valid ? 1.f : 0.f;
    }
  }
  float w00 = (1.f - fx) * (1.f - fy) * vld[0];
  float w10 = fx * (1.f - fy) * vld[1];
  float w01 = (1.f - fx) * fy * vld[2];
  float w11 = fx * fy * vld[3];

  float simsum = 0.f;
  #pragma unroll
  for (int g = 0; g < Gc; ++g) {
    float acc = 0.f;
    #pragma unroll
    for (int c = 0; c < Cc / Gc; ++c) {
      const float* f = src + (g * (Cc / Gc) + c) * HWc;
      float s = w00 * f[idx[0]] + w10 * f[idx[1]] + w01 * f[idx[2]] + w11 * f[idx[3]];
      acc += ref[(g * (Cc / Gc) + c) * HWc + p] * s;
    }
    gsim[(g * Dc + d) * HWc + p] = acc;
    simsum += acc;
  }
  sim[t] = simsum * (1.f / (float)Cc);
}

// ---------------------------------------------------------------------------
// 5) Per-pixel depth MLP (1->16->8->8) fused with max over D -> vmax (8 x HW).
// ---------------------------------------------------------------------------
__global__ void mlp_vmax_kernel(const float* __restrict__ sim,
                                const float* __restrict__ pw0,   // 16
                                const float* __restrict__ pw1,   // 8x16
                                const float* __restrict__ pw2w,  // 8x8
                                const float* __restrict__ pw2b,  // 8
                                float* __restrict__ vmax_out) {  // 8 x HW
  __shared__ float sw[16 + 128 + 64 + 8];
  for (int i = threadIdx.x; i < 16; i += blockDim.x) sw[i] = pw0[i];
  for (int i = threadIdx.x; i < 128; i += blockDim.x) sw[16 + i] = pw1[i];
  for (int i = threadIdx.x; i < 64; i += blockDim.x) sw[144 + i] = pw2w[i];
  for (int i = threadIdx.x; i < 8; i += blockDim.x) sw[208 + i] = pw2b[i];
  __syncthreads();
  int p = blockIdx.x * blockDim.x + threadIdx.x;
  if (p >= HWc) return;
  float vm[8];
  #pragma unroll
  for (int k = 0; k < 8; ++k) vm[k] = -3.402823466e38f;
  for (int d = 0; d < Dc; ++d) {
    float s = sim[d * HWc + p];
    float h0[16];
    #pragma unroll
    for (int j = 0; j < 16; ++j) { float t = sw[j] * s; h0[j] = t > 0.f ? t : 0.f; }
    float h1[8];
    #pragma unroll
    for (int k = 0; k < 8; ++k) {
      float t = 0.f;
      #pragma unroll
      for (int j = 0; j < 16; ++j) t += sw[16 + k * 16 + j] * h0[j];
      h1[k] = t > 0.f ? t : 0.f;
    }
    #pragma unroll
    for (int k = 0; k < 8; ++k) {
      float t = sw[208 + k];
      #pragma unroll
      for (int j = 0; j < 8; ++j) t += sw[144 + k * 8 + j] * h1[j];
      vm[k] = fmaxf(vm[k], t);
    }
  }
  #pragma unroll
  for (int k = 0; k < 8; ++k) vmax_out[k * HWc + p] = vm[k];
}

// ---------------------------------------------------------------------------
// 6) 3x3 conv as implicit-im2col GEMM on V_WMMA_F32_16X16X4_F32.
//    Block = 128 threads = 4 waves; M-tile = 16 row-contiguous pixels
//    (W=160 % 16 == 0, tile never crosses a row); wave w -> out chans
//    [16w,16w+16). Input patch (3 x 18 x CIN) staged in LDS, padded to
//    CP=CIN+2 floats so the 16-lane stride (42/66) hits distinct banks and
//    A pairs are 8B aligned (ds_load_b64). K = (ky*3+kx)*CIN + ci, fully
//    unrolled -> 90/144 back-to-back WMMAs.
//    Fragment layout per ISA 7.12.2 (fp32 16x16x4):
//      A: M = lane&15, K = kk + 2*(lane>>4) + {0,1}
//      B: N = lane&15, K = kk + 2*(lane>>4) + {0,1}
//      C/D: N = lane&15, M = vgpr + 8*(lane>>4)
//    No divergence before the WMMAs -> EXEC all ones as required.
// ---------------------------------------------------------------------------
template <int CIN, int C0>
__global__ __launch_bounds__(128)
void conv3x3_wmma_kernel(const float* __restrict__ in0,
                         const float* __restrict__ in1,
                         const float* __restrict__ bp,   // packed weights
                         float* __restrict__ out) {      // 64 x HW (ReLU)
  constexpr int CP = CIN + 2;
  __shared__ float patch[54 * CP];      // [row*18+col][ci]

  const int tid = threadIdx.x;
  const int p0  = blockIdx.x * 16;
  const int h   = p0 / Wc;
  const int w0  = p0 - h * Wc;

  // Cooperative fill: consecutive tids -> consecutive cols -> coalesced.
  for (int ii = tid; ii < 54 * CIN; ii += 128) {
    int cc = ii / 54;
    int rc = ii - cc * 54;
    int row = rc / 18;
    int col = rc - row * 18;
    int ih = h + row - 1;
    int iw = w0 + col - 1;
    float v = 0.f;
    if (ih >= 0 && ih < Hc && iw >= 0 && iw < Wc) {
      const float* s = (cc < C0) ? (in0 + cc * HWc) : (in1 + (cc - C0) * HWc);
      v = s[ih * Wc + iw];
    }
    patch[rc * CP + cc] = v;
  }
  __syncthreads();

  const int lane = tid & 31;
  const int wave = tid >> 5;
  const int half = lane >> 4;
  const int m    = lane & 15;
  const int n    = wave * 16 + m;       // output channel

  v8f acc = {};
  #pragma unroll
  for (int pos = 0; pos < 9; ++pos) {
    const int ky = pos / 3;             // constants after unroll
    const int kx = pos - ky * 3;
    const float* arow = &patch[(ky * 18 + m + kx) * CP + 2 * half];
    const float* brow = bp + (pos * CIN / 2) * 128 + half * 128 + n * 2;
    #pragma unroll
    for (int cc = 0; cc < CIN; cc += 4) {
      v2f A, Bv;
      A.x  = arow[cc];                  // contiguous pair -> ds_load_b64
      A.y  = arow[cc + 1];
      Bv.x = brow[(cc / 2) * 128];      // contiguous pair -> global b64
      Bv.y = brow[(cc / 2) * 128 + 1];
      acc = __builtin_amdgcn_wmma_f32_16x16x4_f32(false, A, false, Bv,
                                                  (short)0, acc, false, false);
    }
  }
  #pragma unroll
  for (int r = 0; r < 8; ++r) {
    float v = acc[r];
    out[n * HWc + p0 + half * 8 + r] = v > 0.f ? v : 0.f;   // fused ReLU
  }
}

// ---------------------------------------------------------------------------
// 7) Last conv layer (64->1) + ReLU + sigmoid -> view weight.
// ---------------------------------------------------------------------------
__global__ void conv3_sig_kernel(const float* __restrict__ x2,
                                 const float* __restrict__ wgt,  // 1x64x3x3
                                 float* __restrict__ vw) {
  int p = blockIdx.x * blockDim.x + threadIdx.x;
  if (p >= HWc) return;
  int h = p / Wc, w = p - h * Wc;
  float acc = 0.f;
  for (int ci = 0; ci < 64; ++ci) {
    #pragma unroll
    for (int ky = 0; ky < 3; ++ky) {
      int ih = h + ky - 1;
      if (ih < 0 || ih >= Hc) continue;
      #pragma unroll
      for (int kx = 0; kx < 3; ++kx) {
        int iw = w + kx - 1;
        if (iw < 0 || iw >= Wc) continue;
        acc += x2[ci * HWc + ih * Wc + iw] * wgt[(ci * 3 + ky) * 3 + kx];
      }
    }
  }
  acc = acc > 0.f ? acc : 0.f;
  vw[p] = 1.f / (1.f + expf(-acc));
}

// ---------------------------------------------------------------------------
// 8) Weighted accumulation.
// ---------------------------------------------------------------------------
__global__ void accum_kernel(const float* __restrict__ gsim,
                             const float* __restrict__ vw,
                             float* __restrict__ vsum) {
  int t = blockIdx.x * blockDim.x + threadIdx.x;
  if (t >= Gc * Dc * HWc) return;
  int p = t % HWc;
  vsum[t] += gsim[t] * 0.25f * vw[p];   // /(C//G)=4 folded in
}

__global__ void wsum_kernel(const float* __restrict__ vw, float* __restrict__ wsum) {
  int p = blockIdx.x * blockDim.x + threadIdx.x;
  if (p >= HWc) return;
  wsum[p] += vw[p];
}

__global__ void voldiv_kernel(const float* __restrict__ vsum,
                              const float* __restrict__ wsum,
                              float* __restrict__ vol) {
  int t = blockIdx.x * blockDim.x + threadIdx.x;
  if (t >= Gc * Dc * HWc) return;
  vol[t] = vsum[t] / wsum[t % HWc];
}

// ---------------------------------------------------------------------------
// 9) Conv3d 3x3x3, 8 -> 1, pad 1 + bias.
// ---------------------------------------------------------------------------
__global__ void conv3d_kernel(const float* __restrict__ vol,   // 8 x D x HW
                              const float* __restrict__ wgt,   // 1x8x3x3x3
                              const float* __restrict__ bias,
                              float* __restrict__ cost) {      // D x HW
  int t = blockIdx.x * blockDim.x + threadIdx.x;
  if (t >= Dc * HWc) return;
  int d = t / HWc;
  int p = t - d * HWc;
  int h = p / Wc, w = p - h * Wc;
  float acc = bias[0];
  for (int g = 0; g < Gc; ++g) {
    #pragma unroll
    for (int kd = 0; kd < 3; ++kd) {
      int id = d + kd - 1;
      if (id < 0 || id >= Dc) continue;
      #pragma unroll
      for (int kh = 0; kh < 3; ++kh) {
        int ih = h + kh - 1;
        if (ih < 0 || ih >= Hc) continue;
        #pragma unroll
        for (int kw = 0; kw < 3; ++kw) {
          int iw = w + kw - 1;
          if (iw < 0 || iw >= Wc) continue;
          acc += vol[(g * Dc + id) * HWc + ih * Wc + iw] *
                 wgt[((g * 3 + kd) * 3 + kh) * 3 + kw];
        }
      }
    }
  }
  cost[t] = acc;
}

// ---------------------------------------------------------------------------
// 10) Softmax over D + expected depth + 4-tap confidence.
// ---------------------------------------------------------------------------
__global__ void softmax_depth_conf_kernel(const float* __restrict__ cost,
                                          const float* __restrict__ dv,
                                          float* __restrict__ depth_o,
                                          float* __restrict__ conf_o,
                                          float* __restrict__ prob_o) {
  int p = blockIdx.x * blockDim.x + threadIdx.x;
  if (p >= HWc) return;
  float mx = -3.402823466e38f;
  for (int d = 0; d < Dc; ++d) mx = fmaxf(mx, cost[d * HWc + p]);
  float ssum = 0.f;
  for (int d = 0; d < Dc; ++d) ssum += expf(cost[d * HWc + p] - mx);
  float inv = 1.f / ssum;
  float dep = 0.f, dif = 0.f;
  for (int d = 0; d < Dc; ++d) {
    float pr = expf(cost[d * HWc + p] - mx) * inv;
    prob_o[d * HWc + p] = pr;
    dep += pr * dv[d];
    dif += pr * (float)d;
  }
  int di = (int)dif;                    // trunc like .astype(int32)
  di = di < 0 ? 0 : (di > Dc - 1 ? Dc - 1 : di);
  float conf = 0.f;
  for (int dd = di - 1; dd <= di + 2; ++dd)
    if (dd >= 0 && dd < Dc) conf += expf(cost[dd * HWc + p] - mx) * inv;
  depth_o[p] = dep;
  conf_o[p] = conf;
}

// ---------------------------------------------------------------------------
// Launcher. Workspace layout (floats):
//   rt[64] | gsim[8*D*HW] | sim[D*HW] | vsum[8*D*HW] | wsum[HW]
//   | vmax[8*HW] | x1[64*HW] | x2[64*HW] | bp3[64*360] | bp4[64*576]
// Output layout: depth[HW] | conf[HW] | prob[D*HW] | view_weights[3*HW]
// ---------------------------------------------------------------------------
extern "C" void kernel_launch(void* const* d_in, const int* in_sizes, int n_in,
                              void* d_out, int out_size, void* d_ws, size_t ws_size,
                              hipStream_t stream) {
  (void)in_sizes; (void)n_in; (void)out_size; (void)ws_size;
  const float* features = (const float*)d_in[0];   // (4,1,32,128,160)
  const float* projm    = (const float*)d_in[1];   // (1,4,2,4,4)
  const float* dv       = (const float*)d_in[2];   // (1,64)
  const float* pw0      = (const float*)d_in[4];
  const float* pw1      = (const float*)d_in[5];
  const float* pw2w     = (const float*)d_in[6];
  const float* pw2b     = (const float*)d_in[7];
  const float* pw3      = (const float*)d_in[8];   // (64,40,3,3)
  const float* pw4      = (const float*)d_in[9];   // (64,64,3,3)
  const float* pw5      = (const float*)d_in[10];  // (1,64,3,3)
  const float* cregw    = (const float*)d_in[11];  // (1,8,3,3,3)
  const float* cregb    = (const float*)d_in[12];  // (1,)

  float* ws    = (float*)d_ws;
  float* rt    = ws;
  float* gsim  = ws + 64;
  float* sim   = gsim + (size_t)Gc * Dc * HWc;     // also reused for cost
  float* vsum  = sim + (size_t)Dc * HWc;
  float* wsum  = vsum + (size_t)Gc * Dc * HWc;
  float* vmaxb = wsum + HWc;
  float* x1    = vmaxb + (size_t)8 * HWc;
  float* x2    = x1 + (size_t)64 * HWc;
  float* bp3   = x2 + (size_t)64 * HWc;            // 64*360
  float* bp4   = bp3 + (size_t)64 * 360;           // 64*576

  float* out     = (float*)d_out;
  float* depth_o = out;
  float* conf_o  = out + HWc;
  float* prob_o  = out + 2 * HWc;
  float* vw_o    = out + 2 * HWc + (size_t)Dc * HWc;

  proj_kernel<<<1, 1, 0, stream>>>(projm, rt);
  init_kernel<<<(Gc * Dc * HWc + 255) / 256, 256, 0, stream>>>(vsum, wsum);
  pack_weights_kernel<<<(64 * 360 + 255) / 256, 256, 0, stream>>>(pw3, bp3, 40);
  pack_weights_kernel<<<(64 * 576 + 255) / 256, 256, 0, stream>>>(pw4, bp4, 64);

  for (int v = 1; v < Vc; ++v) {
    const float* src = features + (size_t)v * Cc * HWc;
    float* vw = vw_o + (size_t)(v - 1) * HWc;

    warp_gsim_kernel<<<(Dc * HWc + 255) / 256, 256, 0, stream>>>(
        features, src, dv, rt + (v - 1) * 12, gsim, sim);
    mlp_vmax_kernel<<<(HWc + 127) / 128, 128, 0, stream>>>(
        sim, pw0, pw1, pw2w, pw2b, vmaxb);
    // conv1: concat(ref_fea[32], vmax[8]) -> 64, fp32 WMMA
    conv3x3_wmma_kernel<40, 32><<<HWc / 16, 128, 0, stream>>>(
        features, vmaxb, bp3, x1);
    // conv2: 64 -> 64, fp32 WMMA
    conv3x3_wmma_kernel<64, 64><<<HWc / 16, 128, 0, stream>>>(
        x1, nullptr, bp4, x2);
    conv3_sig_kernel<<<(HWc + 255) / 256, 256, 0, stream>>>(x2, pw5, vw);
    accum_kernel<<<(Gc * Dc * HWc + 255) / 256, 256, 0, stream>>>(gsim, vw, vsum);
    wsum_kernel<<<(HWc + 255) / 256, 256, 0, stream>>>(vw, wsum);
  }

  voldiv_kernel<<<(Gc * Dc * HWc + 255) / 256, 256, 0, stream>>>(vsum, wsum, gsim);
  conv3d_kernel<<<(Dc * HWc + 255) / 256, 256, 0, stream>>>(gsim, cregw, cregb, sim);
  softmax_depth_conf_kernel<<<(HWc + 255) / 256, 256, 0, stream>>>(
      sim, dv, depth_o, conf_o, prob_o);
}